// RefinementDecoder_74028056314376
// MI455X (gfx1250) — compile-verified
//
#include <hip/hip_runtime.h>
#include <hip/hip_bf16.h>

// ---------------- problem constants (match reference) ----------------
#define BB   64
#define SS   400
#define HH   1024
#define EE   300
#define VV   50000
#define NOOV 50
#define VEXT (VV + NOOV)          // 50050
#define UNK_ID 1
#define NEGINF (-1.0e12f)

typedef __attribute__((ext_vector_type(2))) float v2f;
typedef __attribute__((ext_vector_type(8))) float v8f;

__device__ __forceinline__ float sigm(float x) { return 1.0f / (1.0f + expf(-x)); }

// monotonic float<->uint key for atomicMax on floats
__device__ __forceinline__ unsigned fkey(float f) {
  unsigned u = __float_as_uint(f);
  return (u & 0x80000000u) ? ~u : (u | 0x80000000u);
}
__device__ __forceinline__ float funkey(unsigned k) {
  unsigned u = (k & 0x80000000u) ? (k & 0x7FFFFFFFu) : ~k;
  return __uint_as_float(u);
}

// ---------------------------------------------------------------------
// WMMA f32 GEMM: C(64 x N) = A(64 x K) @ B(K x N) [+bias or +=C]
// One wave computes a 64x16 strip: 4 M-tiles share every B load.
// V_WMMA_F32_16X16X4_F32 (wave32). Requires N%16==0, K%4==0.
// Inner loop: stage a 16-deep K chunk of operands (one load clause, one
// loadcnt wait), then 16 back-to-back WMMAs on 4 independent acc chains.
// TRANSB: B stored N x K.  ACCUM: C += A@B.  NTB: non-temporal B loads
// (stream 205MB W_logit past L2 so the reused enc_out set stays resident).
// ---------------------------------------------------------------------
template <bool TRANSB, bool ACCUM, bool NTB>
__global__ void wmma_gemm64_f32(const float* __restrict__ A, int lda,
                                const float* __restrict__ Bm, int ldb,
                                const float* __restrict__ bias,
                                float* __restrict__ C, int ldc,
                                int N, int K) {
  const int lane = threadIdx.x & 31;
  const int wave = threadIdx.x >> 5;
  const int tile = blockIdx.x * (blockDim.x >> 5) + wave;
  const int ntiles = N >> 4;
  if (tile >= ntiles) return;                 // wave-uniform exit (EXEC all-1 for WMMA)
  const int nBase = tile << 4;
  const int half  = lane >> 4;                // 0: lanes 0-15, 1: lanes 16-31
  const int l15   = lane & 15;
  const int ncol  = nBase + l15;

  v8f acc[4];
  if (ACCUM) {
#pragma unroll
    for (int mt = 0; mt < 4; ++mt)
#pragma unroll
      for (int r = 0; r < 8; ++r)
        acc[mt][r] = C[(size_t)(mt * 16 + half * 8 + r) * ldc + ncol];
  } else {
    float bv = bias ? bias[ncol] : 0.0f;
#pragma unroll
    for (int mt = 0; mt < 4; ++mt)
#pragma unroll
      for (int r = 0; r < 8; ++r) acc[mt][r] = bv;
  }

  const float* pA = A + (size_t)l15 * lda + half * 2;             // + mt*16*lda
  const float* pB = TRANSB ? (Bm + (size_t)ncol * ldb + half * 2)
                           : (Bm + (size_t)(half * 2) * ldb + ncol);
  const size_t aOff = (size_t)16 * lda;
  const size_t bStep = TRANSB ? (size_t)4 : (size_t)4 * ldb;

  int k = 0;
  // main loop: 16-deep K chunks, operands staged before compute
  for (; k + 16 <= K; k += 16) {
    v2f bb[4];
    v2f aa[4][4];
#pragma unroll
    for (int q = 0; q < 4; ++q) {
      if (TRANSB) {
        bb[q][0] = pB[0];
        bb[q][1] = pB[1];
      } else if (NTB) {
        bb[q][0] = __builtin_nontemporal_load(pB);
        bb[q][1] = __builtin_nontemporal_load(pB + ldb);
      } else {
        bb[q][0] = pB[0];
        bb[q][1] = pB[ldb];
      }
      pB += bStep;
#pragma unroll
      for (int mt = 0; mt < 4; ++mt) {
        aa[q][mt][0] = pA[mt * aOff + 0];
        aa[q][mt][1] = pA[mt * aOff + 1];
      }
      pA += 4;
    }
#pragma unroll
    for (int q = 0; q < 4; ++q)
#pragma unroll
      for (int mt = 0; mt < 4; ++mt)
        acc[mt] = __builtin_amdgcn_wmma_f32_16x16x4_f32(
            false, aa[q][mt], false, bb[q], (short)0, acc[mt], false, false);
  }
  // tail: 4-deep steps (covers K % 16, e.g. K = 300)
  for (; k < K; k += 4) {
    v2f b;
    if (TRANSB) {
      b[0] = pB[0];
      b[1] = pB[1];
    } else {
      b[0] = pB[0];
      b[1] = pB[ldb];
    }
    pB += bStep;
#pragma unroll
    for (int mt = 0; mt < 4; ++mt) {
      v2f a;
      a[0] = pA[mt * aOff + 0];
      a[1] = pA[mt * aOff + 1];
      acc[mt] = __builtin_amdgcn_wmma_f32_16x16x4_f32(
          false, a, false, b, (short)0, acc[mt], false, false);
    }
    pA += 4;
  }

#pragma unroll
  for (int mt = 0; mt < 4; ++mt)
#pragma unroll
    for (int r = 0; r < 8; ++r)
      C[(size_t)(mt * 16 + half * 8 + r) * ldc + ncol] = acc[mt][r];
}

// ------- pooled_raw[b,h] = mean_s enc[b,s,h]  (float4-vectorized stream) -------
__global__ void mean_s_kernel(const float* __restrict__ enc, float* __restrict__ pooled) {
  int idx = blockIdx.x * blockDim.x + threadIdx.x;   // B*H/4 threads
  int b = idx >> 8, kq = idx & (HH / 4 - 1);
  const float4* p = (const float4*)(enc + (size_t)b * SS * HH) + kq;
  float4 s = make_float4(0.f, 0.f, 0.f, 0.f);
  for (int si = 0; si < SS; ++si) {
    float4 v = p[(size_t)si * (HH / 4)];
    s.x += v.x; s.y += v.y; s.z += v.z; s.w += v.w;
  }
  const float inv = 1.0f / SS;
  s.x *= inv; s.y *= inv; s.z *= inv; s.w *= inv;
  ((float4*)pooled)[idx] = s;
}

// ------- combined = [emb[y] | enc_pooled | decoder_out] @ W_comb + b -------
__global__ void combined_kernel(const int* __restrict__ y, const float* __restrict__ emb,
                                const float* __restrict__ encp, const float* __restrict__ dec,
                                const float* __restrict__ W, const float* __restrict__ bias,
                                float* __restrict__ out) {
  int idx = blockIdx.x * blockDim.x + threadIdx.x;   // B*E threads
  int b = idx / EE, j = idx - b * EE;
  float s = bias[j];
  const float* er = emb + (size_t)y[b] * EE;
  for (int k = 0; k < EE; ++k) s += er[k] * W[(size_t)k * EE + j];
  const float* pp = encp + b * HH;
  for (int k = 0; k < HH; ++k) s += pp[k] * W[(size_t)(EE + k) * EE + j];
  const float* dd = dec + b * EE;
  for (int k = 0; k < EE; ++k) s += dd[k] * W[(size_t)(EE + HH + k) * EE + j];
  out[idx] = s;
}

// ------- lstm_in = [combined | prev_context] @ W_red + b_red -------
__global__ void lstmin_kernel(const float* __restrict__ comb, const float* __restrict__ pctx,
                              const float* __restrict__ W, const float* __restrict__ bias,
                              float* __restrict__ out) {
  int idx = blockIdx.x * blockDim.x + threadIdx.x;   // B*E threads
  int b = idx / EE, j = idx - b * EE;
  float s = bias[j];
  const float* cc = comb + b * EE;
  for (int k = 0; k < EE; ++k) s += cc[k] * W[(size_t)k * EE + j];
  const float* pc = pctx + b * HH;
  for (int k = 0; k < HH; ++k) s += pc[k] * W[(size_t)(EE + k) * EE + j];
  out[idx] = s;
}

// ---------------- LSTM pointwise ----------------
__global__ void lstm_kernel(const float* __restrict__ gates, const float* __restrict__ b_hh,
                            const float* __restrict__ prev_c,
                            float* __restrict__ h_out, float* __restrict__ c_out) {
  int idx = blockIdx.x * blockDim.x + threadIdx.x;   // B*H threads
  int b = idx >> 10, j = idx & (HH - 1);
  const float* g = gates + (size_t)b * 4 * HH;
  float gi = g[j] + b_hh[j];
  float gf = g[HH + j] + b_hh[HH + j];
  float gg = g[2 * HH + j] + b_hh[2 * HH + j];
  float go = g[3 * HH + j] + b_hh[3 * HH + j];
  float c = sigm(gf) * prev_c[idx] + sigm(gi) * tanhf(gg);
  float h = sigm(go) * tanhf(c);
  h_out[idx] = h;
  c_out[idx] = c;
}

// ---------------- hdotb[b] = b_enc . h[b] ----------------
__global__ void hdot_kernel(const float* __restrict__ b_enc, const float* __restrict__ h,
                            float* __restrict__ hd) {
  int b = blockIdx.x, lane = threadIdx.x;            // 32 threads/block
  float s = 0.0f;
  for (int j = lane; j < HH; j += 32) s += b_enc[j] * h[b * HH + j];
  for (int o = 16; o; o >>= 1) s += __shfl_xor(s, o, 32);
  if (lane == 0) hd[b] = s;
}

// ------- energy[b,s] = enc[b,s,:].t[b,:] + hdotb[b], masked to -INF -------
// one wave per (b,s); float4 (b128) loads, 8 iterations per lane
__global__ void energy_kernel(const float* __restrict__ enc, const float* __restrict__ t,
                              const float* __restrict__ hdotb, const int* __restrict__ mask,
                              float* __restrict__ energy) {
  int wid  = (blockIdx.x * blockDim.x + threadIdx.x) >> 5;
  int lane = threadIdx.x & 31;
  int b = wid / SS;
  const float4* e4 = (const float4*)(enc + (size_t)wid * HH);
  const float4* t4 = (const float4*)(t + b * HH);
  float acc = 0.0f;
#pragma unroll
  for (int k = lane; k < HH / 4; k += 32) {
    float4 ev = e4[k], tv = t4[k];
    acc += ev.x * tv.x + ev.y * tv.y + ev.z * tv.z + ev.w * tv.w;
  }
  for (int o = 16; o; o >>= 1) acc += __shfl_xor(acc, o, 32);
  if (lane == 0) energy[wid] = mask[wid] ? NEGINF : (acc + hdotb[b]);
}

// ---------------- softmax over S per batch row ----------------
__global__ void softmax_kernel(const float* __restrict__ energy, float* __restrict__ attn) {
  __shared__ float red[256];
  int b = blockIdx.x, tid = threadIdx.x;
  const float* e = energy + b * SS;
  float m = -3.4e38f;
  for (int s = tid; s < SS; s += 256) m = fmaxf(m, e[s]);
  red[tid] = m; __syncthreads();
  for (int o = 128; o; o >>= 1) { if (tid < o) red[tid] = fmaxf(red[tid], red[tid + o]); __syncthreads(); }
  m = red[0]; __syncthreads();
  float sum = 0.0f;
  for (int s = tid; s < SS; s += 256) sum += expf(e[s] - m);
  red[tid] = sum; __syncthreads();
  for (int o = 128; o; o >>= 1) { if (tid < o) red[tid] += red[tid + o]; __syncthreads(); }
  float inv = 1.0f / red[0];
  for (int s = tid; s < SS; s += 256) attn[b * SS + s] = expf(e[s] - m) * inv;
}

// ------- araw[b,k4] = sum_s attn[b,s]*enc[b,s,k4]  (float4-vectorized) -------
__global__ void attnmix_kernel(const float* __restrict__ attn, const float* __restrict__ enc,
                               float* __restrict__ araw) {
  int idx = blockIdx.x * blockDim.x + threadIdx.x;   // B*H/4 threads
  int b = idx >> 8, kq = idx & (HH / 4 - 1);
  const float* a = attn + b * SS;
  const float4* e = (const float4*)(enc + (size_t)b * SS * HH) + kq;
  float4 s = make_float4(0.f, 0.f, 0.f, 0.f);
  for (int si = 0; si < SS; ++si) {
    float w = a[si];
    float4 v = e[(size_t)si * (HH / 4)];
    s.x += w * v.x; s.y += w * v.y; s.z += w * v.z; s.w += w * v.w;
  }
  ((float4*)araw)[idx] = s;
}

__global__ void tanh_kernel(float* __restrict__ x, int n) {
  int i = blockIdx.x * blockDim.x + threadIdx.x;
  if (i < n) x[i] = tanhf(x[i]);
}

// zero the OOV logit columns and initialize scatter keys to key(-INF)
__global__ void prep_kernel(float* __restrict__ out, unsigned* __restrict__ keys, int total) {
  int idx = blockIdx.x * blockDim.x + threadIdx.x;
  if (idx >= total) return;
  int v = idx % VEXT;
  if (v >= VV) out[idx] = 0.0f;
  keys[idx] = fkey(NEGINF);
}

__global__ void scatter_kernel(const int* __restrict__ ext_x, const float* __restrict__ energy,
                               unsigned* __restrict__ keys) {
  int idx = blockIdx.x * blockDim.x + threadIdx.x;   // B*S threads
  int b = idx / SS;
  int v = ext_x[idx];
  atomicMax(&keys[(size_t)b * VEXT + v], fkey(energy[idx]));
}

__global__ void finalize_kernel(float* __restrict__ out, const unsigned* __restrict__ keys,
                                int total) {
  int idx = blockIdx.x * blockDim.x + threadIdx.x;
  if (idx >= total) return;
  int v = idx % VEXT;
  float base = out[idx];
  float sv = funkey(keys[idx]);
  float sc = (sv == NEGINF) ? 0.0f : sv;
  float o = base + sc;
  if (o == NEGINF) o = 0.0f;
  if (v == UNK_ID) o = NEGINF;
  out[idx] = o;
}

// ---------------------------------------------------------------------
extern "C" void kernel_launch(void* const* d_in, const int* in_sizes, int n_in,
                              void* d_out, int out_size, void* d_ws, size_t ws_size,
                              hipStream_t stream) {
  const int*   y       = (const int*)  d_in[0];
  const int*   ext_x   = (const int*)  d_in[1];
  const float* prev_h  = (const float*)d_in[2];
  const float* prev_c  = (const float*)d_in[3];
  const float* prev_ctx= (const float*)d_in[4];
  const float* enc     = (const float*)d_in[5];
  const int*   mask    = (const int*)  d_in[6];
  const float* dec     = (const float*)d_in[7];
  const float* emb     = (const float*)d_in[8];
  const float* W_enc   = (const float*)d_in[9];
  const float* b_enc   = (const float*)d_in[10];
  const float* W_comb  = (const float*)d_in[11];
  const float* b_comb  = (const float*)d_in[12];
  const float* W_red   = (const float*)d_in[13];
  const float* b_red   = (const float*)d_in[14];
  const float* W_ih    = (const float*)d_in[15];
  const float* W_hh    = (const float*)d_in[16];
  const float* b_ih    = (const float*)d_in[17];
  const float* b_hh    = (const float*)d_in[18];
  const float* W_cat   = (const float*)d_in[19];
  const float* b_cat   = (const float*)d_in[20];
  const float* W_logit = (const float*)d_in[21];
  const float* b_logit = (const float*)d_in[22];

  float* out_logit = (float*)d_out;                       // (B, VEXT)
  float* h_out   = out_logit + (size_t)BB * VEXT;         // (B, H)
  float* c_out   = h_out + BB * HH;                       // (B, H)
  float* ctx_out = c_out + BB * HH;                       // (B, H)

  // workspace layout (floats)
  float* ws      = (float*)d_ws;
  float* pooled  = ws;                    // B*H
  float* encp    = pooled + BB * HH;      // B*H
  float* comb    = encp + BB * HH;        // B*E
  float* lin     = comb + BB * EE;        // B*E
  float* gates   = lin + BB * EE;         // B*4H
  float* tvec    = gates + BB * 4 * HH;   // B*H
  float* hdotb   = tvec + BB * HH;        // B
  float* energy  = hdotb + BB;            // B*S
  float* attn    = energy + BB * SS;      // B*S
  float* araw    = attn + BB * SS;        // B*H
  float* li      = araw + BB * HH;        // B*H
  unsigned* keys = (unsigned*)(li + BB * HH); // B*VEXT

  const int total = BB * VEXT;
  auto gemmGrid = [](int N) { return dim3(((N >> 4) + 7) / 8); };

  // 1) pooled_raw = mean_s(enc)
  mean_s_kernel<<<BB * HH / 4 / 256, 256, 0, stream>>>(enc, pooled);
  // 2) enc_pooled = pooled_raw @ W_enc + b_enc   (mean commutes with linear)
  wmma_gemm64_f32<false, false, false><<<gemmGrid(HH), 256, 0, stream>>>(
      pooled, HH, W_enc, HH, b_enc, encp, HH, HH, HH);
  // 3) combined, 4) lstm_in
  combined_kernel<<<BB * EE / 256, 256, 0, stream>>>(y, emb, encp, dec, W_comb, b_comb, comb);
  lstmin_kernel<<<BB * EE / 256, 256, 0, stream>>>(comb, prev_ctx, W_red, b_red, lin);
  // 5) gates = lstm_in @ W_ih + b_ih ; gates += h0 @ W_hh
  wmma_gemm64_f32<false, false, false><<<gemmGrid(4 * HH), 256, 0, stream>>>(
      lin, EE, W_ih, 4 * HH, b_ih, gates, 4 * HH, 4 * HH, EE);
  wmma_gemm64_f32<false, true, false><<<gemmGrid(4 * HH), 256, 0, stream>>>(
      prev_h, HH, W_hh, 4 * HH, nullptr, gates, 4 * HH, 4 * HH, HH);
  // 6) LSTM cell -> h, c (written straight to d_out slots)
  lstm_kernel<<<BB * HH / 256, 256, 0, stream>>>(gates, b_hh, prev_c, h_out, c_out);
  // 7) t = h @ W_enc^T  (so energy = enc.t + b_enc.h, never materializing enc_feat)
  wmma_gemm64_f32<true, false, false><<<gemmGrid(HH), 256, 0, stream>>>(
      h_out, HH, W_enc, HH, nullptr, tvec, HH, HH, HH);
  hdot_kernel<<<BB, 32, 0, stream>>>(b_enc, h_out, hdotb);
  // 8) energy + mask
  energy_kernel<<<BB * SS * 32 / 256, 256, 0, stream>>>(enc, tvec, hdotb, mask, energy);
  // 9) softmax
  softmax_kernel<<<BB, 256, 0, stream>>>(energy, attn);
  // 10) context = (attn @ enc) @ W_enc + b_enc
  attnmix_kernel<<<BB * HH / 4 / 256, 256, 0, stream>>>(attn, enc, araw);
  wmma_gemm64_f32<false, false, false><<<gemmGrid(HH), 256, 0, stream>>>(
      araw, HH, W_enc, HH, b_enc, ctx_out, HH, HH, HH);
  // 11) logit_in = tanh([h|context] @ W_cat + b_cat)  (split-K accumulate)
  wmma_gemm64_f32<false, false, false><<<gemmGrid(HH), 256, 0, stream>>>(
      h_out, HH, W_cat, HH, b_cat, li, HH, HH, HH);
  wmma_gemm64_f32<false, true, false><<<gemmGrid(HH), 256, 0, stream>>>(
      ctx_out, HH, W_cat + (size_t)HH * HH, HH, nullptr, li, HH, HH, HH);
  tanh_kernel<<<BB * HH / 256, 256, 0, stream>>>(li, BB * HH);
  // 12) prep scat keys + zero OOV columns (touches only cols >= V of d_out)
  prep_kernel<<<(total + 255) / 256, 256, 0, stream>>>(out_logit, keys, total);
  // 13) logit = logit_in @ W_logit + b_logit -> d_out (ldc=VEXT); NT loads keep
  //     the streamed 205MB W_logit from evicting enc_out from L2
  wmma_gemm64_f32<false, false, true><<<gemmGrid(VV), 256, 0, stream>>>(
      li, HH, W_logit, VV, b_logit, out_logit, VEXT, VV, HH);
  // 14) scatter-max of (masked) energy into extended vocab, then finalize
  scatter_kernel<<<BB * SS / 256, 256, 0, stream>>>(ext_x, energy, keys);
  finalize_kernel<<<(total + 255) / 256, 256, 0, stream>>>(out_logit, keys, total);
}